// CustomSTFT_40226663694339
// MI455X (gfx1250) — compile-verified
//
#include <hip/hip_runtime.h>
#include <math.h>

#define NFFT   800
#define HOP    200
#define FREQ   401
#define PADL   400
#define BATCH  32
#define T_IN   480000
#define FRAMES 2401
#define OUTW_PAD 480800          // (FRAMES-1)*HOP + NFFT
#define LDSA_STRIDE 804          // 800 + 4 pad: 16B-aligned rows, banks spread
#define DSTRIDE 20               // D staging row pitch (floats), 16B-aligned
#define PI_F 3.14159265358979323846f

typedef __attribute__((ext_vector_type(2))) float v2f;
typedef __attribute__((ext_vector_type(8))) float v8f;
typedef __attribute__((ext_vector_type(4))) int   v4i;

// ---------------------------------------------------------------------------
// Kernel 1: build the fused STFT->ISTFT round-trip matrix
//   W[w][w'] = win[w]/FREQ * sum_f sc_f * cos(2*pi*f*(w-w')/NFFT)
// ---------------------------------------------------------------------------
__global__ void build_w(float* __restrict__ W) {
  int idx = blockIdx.x * blockDim.x + threadIdx.x;
  if (idx >= NFFT * NFFT) return;
  int w  = idx / NFFT;
  int wp = idx - w * NFFT;
  int d  = w - wp;
  int dr = ((d % NFFT) + NFFT) % NFFT;
  const float twopi_over_n = 2.0f * PI_F / (float)NFFT;
  float s = 0.0f;
  for (int f = 0; f < FREQ; ++f) {
    int rf = (dr * f) % NFFT;
    float c = __cosf(twopi_over_n * (float)rf);
    if (f == 0 || f == FREQ - 1) c *= 0.5f;
    s += c;
  }
  float win = 0.5f - 0.5f * __cosf(2.0f * PI_F * (float)w / (float)(NFFT - 1));
  W[(size_t)w * NFFT + wp] = s * win / (float)FREQ;
}

// ---------------------------------------------------------------------------
// Kernel 2: per-parity-class fused GEMM + overlap-add.
// A-tile staged via GLOBAL_LOAD_ASYNC_TO_LDS_B128 (ASYNCcnt path); reflect
// edges / tail rows fall back to scalar fills.  8 wave32s; each wave owns
// pairs of adjacent 16-wide N-tiles (2 accumulators per A load).
// ---------------------------------------------------------------------------
template <bool INIT>
__global__ void __launch_bounds__(256)
istft_pass(const float* __restrict__ x,      // [BATCH][T_IN]
           const float* __restrict__ Wm,     // [NFFT][NFFT]
           float* __restrict__ acc,          // [BATCH][OUTW_PAD]
           int cls)
{
  __shared__ float Alds[16 * LDSA_STRIDE];      // 16 frames x 800 samples
  __shared__ float Dlds[8][16][DSTRIDE];        // per-wave D-tile staging

  const int b    = blockIdx.y;
  const int tile = blockIdx.x;
  const int tid  = threadIdx.x;

  // ---- Stage A-tile: async global->LDS b128 fast path ----
  const float* xb = x + (size_t)b * T_IN;
  for (int u = tid; u < 16 * NFFT / 4; u += 256) {
    int base = u * 4;                       // element index in 16x800 tile
    int m = base / NFFT;
    int w = base - m * NFFT;                // multiple of 4; rows are 800%4==0
    int t = cls + 4 * (tile * 16 + m);
    float* dstp = &Alds[m * LDSA_STRIDE + w];
    if (t < FRAMES) {
      int src0 = t * HOP + w - PADL;
      if (src0 >= 0 && src0 + 3 < T_IN) {
        // contiguous, 16B-aligned on both sides -> async DMA into LDS
        __builtin_amdgcn_global_load_async_to_lds_b128(
            (v4i*)(xb + src0), (v4i*)dstp, 0, 0);
      } else {
        #pragma unroll
        for (int e = 0; e < 4; ++e) {       // reflect-padded edge chunk
          int s2 = src0 + e;
          if (s2 < 0)     s2 = -s2;
          if (s2 >= T_IN) s2 = 2 * (T_IN - 1) - s2;
          dstp[e] = xb[s2];
        }
      }
    } else {
      *(float4*)dstp = float4{0.f, 0.f, 0.f, 0.f};   // tail frame rows
    }
  }
  __builtin_amdgcn_s_wait_asynccnt(0);
  __syncthreads();

  const int lane = tid & 31;
  const int wid  = tid >> 5;
  const int mlo  = lane & 15;          // A row / B,D column within tile
  const int koff = (lane >> 4) * 2;    // K sub-offset for A/B operands
  const int rhi  = (lane >> 4) * 8;    // D row offset for upper half-wave

  const float* Aptr = &Alds[mlo * LDSA_STRIDE + koff];
  float* accb = acc + (size_t)b * OUTW_PAD;

  // 25 pairs of adjacent N-tiles round-robined over the 8 waves.
  for (int p = wid; p < NFFT / 32; p += 8) {
    const int nbase = p * 32;
    const float* Bptr0 = Wm + (size_t)koff * NFFT + nbase + mlo;
    const float* Bptr1 = Bptr0 + 16;
    v8f c0 = {};
    v8f c1 = {};
    #pragma unroll 4
    for (int k = 0; k < NFFT; k += 4) {
      v2f a = *(const v2f*)(Aptr + k);              // shared by both tiles
      v2f b0, b1;
      b0.x = Bptr0[0];     b0.y = Bptr0[NFFT];
      b1.x = Bptr1[0];     b1.y = Bptr1[NFFT];
      Bptr0 += 4 * NFFT;   Bptr1 += 4 * NFFT;
      c0 = __builtin_amdgcn_wmma_f32_16x16x4_f32(false, a, false, b0,
                                                 (short)0, c0, false, false);
      c1 = __builtin_amdgcn_wmma_f32_16x16x4_f32(false, a, false, b1,
                                                 (short)0, c1, false, false);
    }

    // ---- Epilogue: stage D through LDS, then coalesced float4 RMW ----
    const int row  = lane >> 1;
    const int col0 = (lane & 1) * 8;
    const int trow = cls + 4 * (tile * 16 + row);
    #pragma unroll
    for (int half = 0; half < 2; ++half) {
      #pragma unroll
      for (int r = 0; r < 8; ++r)
        Dlds[wid][r + rhi][mlo] = (half ? c1[r] : c0[r]);

      if (trow < FRAMES) {
        const float4 d0 = *(const float4*)&Dlds[wid][row][col0];
        const float4 d1 = *(const float4*)&Dlds[wid][row][col0 + 4];
        float* dst = accb + (size_t)trow * HOP + nbase + half * 16 + col0;
        if (INIT) {
          *(float4*)(dst)     = d0;
          *(float4*)(dst + 4) = d1;
        } else {
          float4 o0 = *(const float4*)(dst);
          float4 o1 = *(const float4*)(dst + 4);
          o0.x += d0.x; o0.y += d0.y; o0.z += d0.z; o0.w += d0.w;
          o1.x += d1.x; o1.y += d1.y; o1.z += d1.z; o1.w += d1.w;
          *(float4*)(dst)     = o0;
          *(float4*)(dst + 4) = o1;
        }
      }
    }
  }
}

// ---------------------------------------------------------------------------
// Kernel 3: strip reflect padding -> final output [BATCH][1][T_IN]
// ---------------------------------------------------------------------------
__global__ void crop_out(const float* __restrict__ acc, float* __restrict__ out) {
  int i = blockIdx.x * blockDim.x + threadIdx.x;
  if (i >= BATCH * T_IN) return;
  int b = i / T_IN;
  int t = i - b * T_IN;
  out[i] = acc[(size_t)b * OUTW_PAD + PADL + t];
}

// ---------------------------------------------------------------------------
extern "C" void kernel_launch(void* const* d_in, const int* in_sizes, int n_in,
                              void* d_out, int out_size, void* d_ws, size_t ws_size,
                              hipStream_t stream) {
  const float* x = (const float*)d_in[0];
  float* out = (float*)d_out;

  float* Wm  = (float*)d_ws;                                        // 2.56 MB
  float* acc = (float*)((char*)d_ws +
                        (size_t)NFFT * NFFT * sizeof(float));       // 61.5 MB

  build_w<<<(NFFT * NFFT + 255) / 256, 256, 0, stream>>>(Wm);

  dim3 grid(38, BATCH);   // ceil(601/16) frame tiles per parity class
  istft_pass<true><<<grid, 256, 0, stream>>>(x, Wm, acc, 0);
  for (int cls = 1; cls < 4; ++cls)
    istft_pass<false><<<grid, 256, 0, stream>>>(x, Wm, acc, cls);

  crop_out<<<(BATCH * T_IN + 255) / 256, 256, 0, stream>>>(acc, out);
}